// MLPLocalCrossAttention_1589137900290
// MI455X (gfx1250) — compile-verified
//
#include <hip/hip_runtime.h>
#include <hip/hip_bf16.h>

// ---------------------------------------------------------------------------
// MLPLocalCrossAttention for MI455X (gfx1250, wave32)
//   inputs : A1[B,N], A2[B,N], wq[16], bq[16], wk[16], bk[16], wv[16], bv[16]
//   outputs: updated_A2[B,N] ++ attn[B,50,50]   (float32, concatenated)
// ---------------------------------------------------------------------------

#define B_ROWS 2048
#define N_COLS 50000
#define K_TOP  50
#define HID    16

typedef __attribute__((ext_vector_type(2))) float v2f;
typedef __attribute__((ext_vector_type(8))) float v8f;

// ------------------------- phase 0: zero fill ------------------------------
__global__ void zero_f4_kernel(float4* __restrict__ p, long long n4) {
    long long i = (long long)blockIdx.x * blockDim.x + threadIdx.x;
    if (i < n4) p[i] = make_float4(0.f, 0.f, 0.f, 0.f);
}

// ------------------------- phase 1: per-row top-k --------------------------
// One 1024-thread workgroup per row; row (200 KB) cached in dynamic LDS
// (CDNA5 WGP has 320 KB LDS).  Row staging uses the CDNA5 async global->LDS
// path (GLOBAL_LOAD_ASYNC_TO_LDS_B128, tracked by ASYNCcnt) which bypasses
// VGPRs.  Then 50 iterations of block-wide argmax with a per-thread 64-bit
// removed mask.  Tie-break: lower global index wins.
__global__ __launch_bounds__(1024) void topk_kernel(
    const float* __restrict__ A1, const float* __restrict__ A2,
    float* __restrict__ tk1, float* __restrict__ tk2, int* __restrict__ idx2)
{
    extern __shared__ float smem[];
    float* s_vals = smem;                    // [N_COLS] row cache (LDS offset 0)
    float* s_rval = smem + 50016;            // [32] per-wave max value
    int*   s_ridx = (int*)(smem + 50048);    // [32] per-wave max index
    int*   s_bidx = (int*)(smem + 50080);    // [1]  winner broadcast

    const int  blk   = blockIdx.x;
    const bool isA2  = (blk >= B_ROWS);
    const int  row   = isA2 ? (blk - B_ROWS) : blk;
    const float* rowp = (isA2 ? A2 : A1) + (size_t)row * N_COLS;
    const int  tid  = threadIdx.x;
    const int  wave = tid >> 5;
    const int  lane = tid & 31;

    // ---- stage row into LDS: 12500 x 16B async copies (one HBM pass) ----
    // Row byte offset = row*200000, multiple of 16 -> b128 aligned.
    // Only dynamic LDS in this kernel, so s_vals starts at LDS offset 0 and
    // the VDST operand (LDS byte address within the wave's allocation) is q*16.
    {
        const float4* rowp4 = (const float4*)rowp;
        for (int q = tid; q < (N_COLS / 4); q += 1024) {
            unsigned lds_byte = (unsigned)(q * 16);
            const float4* gptr = rowp4 + q;
            asm volatile("global_load_async_to_lds_b128 %0, %1, off"
                         :: "v"(lds_byte), "v"(gptr)
                         : "memory");
        }
        asm volatile("s_wait_asynccnt 0" ::: "memory");
    }
    __syncthreads();

    unsigned long long removed = 0ull;

    for (int t = 0; t < K_TOP; ++t) {
        // per-thread scan of its 49 strided elements (LDS, conflict-free)
        float best = -__builtin_inff();
        int   bidx = 0x7fffffff;
        #pragma unroll
        for (int i = 0; i < 49; ++i) {
            int idx = tid + (i << 10);
            if (idx < N_COLS && !((removed >> i) & 1ull)) {
                float v = s_vals[idx];
                if (v > best) { best = v; bidx = idx; }   // keeps lowest idx on tie
            }
        }
        // wave32 reduce (max, tie -> lower index)
        #pragma unroll
        for (int off = 16; off; off >>= 1) {
            float ov = __shfl_xor(best, off, 32);
            int   oi = __shfl_xor(bidx, off, 32);
            if (ov > best || (ov == best && oi < bidx)) { best = ov; bidx = oi; }
        }
        if (lane == 0) { s_rval[wave] = best; s_ridx[wave] = bidx; }
        __syncthreads();
        // cross-wave reduce in wave 0 (32 waves -> 32 lanes)
        if (wave == 0) {
            float v  = s_rval[lane];
            int   ix = s_ridx[lane];
            #pragma unroll
            for (int off = 16; off; off >>= 1) {
                float ov = __shfl_xor(v, off, 32);
                int   oi = __shfl_xor(ix, off, 32);
                if (ov > v || (ov == v && oi < ix)) { v = ov; ix = oi; }
            }
            if (lane == 0) {
                *s_bidx = ix;
                if (isA2) { tk2[row * K_TOP + t] = v; idx2[row * K_TOP + t] = ix; }
                else      { tk1[row * K_TOP + t] = v; }
            }
        }
        __syncthreads();
        int wix = *s_bidx;
        if ((wix & 1023) == tid) removed |= 1ull << (wix >> 10);
        // next iteration's first barrier orders the *s_bidx read vs. rewrite
    }
}

// ------------------------- phase 2: attention ------------------------------
// One 128-thread (4-wave) block per row.  Wave w owns score rows [16w,16w+16).
// scores tile = sum of 4 chained V_WMMA_F32_16X16X4_F32 over HID=16 (exact f32).
// A fragment (16x4): lane = M, vgpr0/1 hold K = {0,1} (lanes 0-15) or {2,3}.
// B fragment (4x16): lane = N, vgpr0/1 hold K = {0,1} / {2,3} (mirrored).
// C/D tile: VGPR r, lanes 0-15 -> M=r, lanes 16-31 -> M=r+8; N = lane%16.
// attn@V followed by mean over h collapses exactly to
//   u_i = (mean wv) * sum_j attn[i,j]*k_j + (mean bv).
__global__ __launch_bounds__(128) void attn_kernel(
    const float* __restrict__ tk1, const float* __restrict__ tk2,
    const int* __restrict__ idx2,
    const float* __restrict__ wq, const float* __restrict__ bq,
    const float* __restrict__ wk, const float* __restrict__ bk,
    const float* __restrict__ wv, const float* __restrict__ bv,
    float* __restrict__ out_updated, float* __restrict__ out_attn)
{
    __shared__ float sq[64], sk[64];
    __shared__ float swq[HID], sbq[HID], swk[HID], sbk[HID];

    const int row  = blockIdx.x;
    const int tid  = threadIdx.x;
    const int wave = tid >> 5;
    const int lane = tid & 31;
    const int n    = lane & 15;    // column-within-tile / row-within-A-tile
    const int hi   = lane >> 4;    // half-wave select

    if (tid < 64) {
        sq[tid] = (tid < K_TOP) ? tk1[(size_t)row * K_TOP + tid] : 0.f;
        sk[tid] = (tid < K_TOP) ? tk2[(size_t)row * K_TOP + tid] : 0.f;
    }
    if (tid < HID) {
        swq[tid] = wq[tid]; sbq[tid] = bq[tid];
        swk[tid] = wk[tid]; sbk[tid] = bk[tid];
    }
    __syncthreads();

    const int i0 = wave * 16;           // row strip handled by this wave
    const float qv = sq[i0 + n];        // q value for A-matrix row M = n

    // A fragments for the 4 K-chunks (K = h dimension, 16 total)
    v2f afr[4];
    #pragma unroll
    for (int kk = 0; kk < 4; ++kk) {
        int h0 = 4 * kk + 2 * hi;
        v2f a; a.x = qv * swq[h0]     + sbq[h0];
               a.y = qv * swq[h0 + 1] + sbq[h0 + 1];
        afr[kk] = a;
    }

    v8f   c[4];
    float kvt[4];
    #pragma unroll
    for (int t = 0; t < 4; ++t) {
        v8f acc = {0.f, 0.f, 0.f, 0.f, 0.f, 0.f, 0.f, 0.f};
        float kv = sk[t * 16 + n];
        kvt[t] = kv;
        #pragma unroll
        for (int kk = 0; kk < 4; ++kk) {
            int h0 = 4 * kk + 2 * hi;
            v2f b; b.x = kv * swk[h0]     + sbk[h0];
                   b.y = kv * swk[h0 + 1] + sbk[h0 + 1];
            acc = __builtin_amdgcn_wmma_f32_16x16x4_f32(
                    /*neg_a=*/false, afr[kk], /*neg_b=*/false, b,
                    /*c_mod=*/(short)0, acc, /*reuse_a=*/false, /*reuse_b=*/false);
        }
        c[t] = acc;
    }

    // mean of value projection (exact collapse of attn@V then mean over h)
    float wvm = 0.f, bvm = 0.f;
    #pragma unroll
    for (int h = 0; h < HID; ++h) { wvm += wv[h]; bvm += bv[h]; }
    wvm *= (1.f / HID); bvm *= (1.f / HID);

    // mask padded columns j >= 50 before softmax
    #pragma unroll
    for (int t = 0; t < 4; ++t) {
        if (16 * t + n >= K_TOP) {
            #pragma unroll
            for (int r = 0; r < 8; ++r) c[t][r] = -__builtin_inff();
        }
    }

    // per-row softmax + outputs; row M = i0 + r + 8*hi lives in 16 lanes of one
    // half-wave, so shfl_xor 8/4/2/1 reduces both halves independently.
    #pragma unroll
    for (int r = 0; r < 8; ++r) {
        const int i = i0 + r + 8 * hi;

        float m = -__builtin_inff();
        #pragma unroll
        for (int t = 0; t < 4; ++t) m = fmaxf(m, c[t][r]);
        #pragma unroll
        for (int off = 8; off; off >>= 1) m = fmaxf(m, __shfl_xor(m, off, 32));

        float ev[4], s = 0.f;
        #pragma unroll
        for (int t = 0; t < 4; ++t) { float e = __expf(c[t][r] - m); ev[t] = e; s += e; }
        #pragma unroll
        for (int off = 8; off; off >>= 1) s += __shfl_xor(s, off, 32);
        const float inv = 1.f / s;

        float up = 0.f;
        #pragma unroll
        for (int t = 0; t < 4; ++t) {
            float a = ev[t] * inv;
            int   j = 16 * t + n;
            if (i < K_TOP && j < K_TOP)
                out_attn[(size_t)row * (K_TOP * K_TOP) + i * K_TOP + j] = a;
            float g = (j < K_TOP) ? (kvt[t] * wvm + bvm) : 0.f;
            up += a * g;
        }
        #pragma unroll
        for (int off = 8; off; off >>= 1) up += __shfl_xor(up, off, 32);

        if (n == 0 && i < K_TOP) {
            int gi = idx2[(size_t)row * K_TOP + i];
            out_updated[(size_t)row * N_COLS + gi] = up;
        }
    }
}

// ---------------------------------------------------------------------------
extern "C" void kernel_launch(void* const* d_in, const int* in_sizes, int n_in,
                              void* d_out, int out_size, void* d_ws, size_t ws_size,
                              hipStream_t stream) {
    const float* A1 = (const float*)d_in[0];
    const float* A2 = (const float*)d_in[1];
    const float* wq = (const float*)d_in[2];
    const float* bq = (const float*)d_in[3];
    const float* wk = (const float*)d_in[4];
    const float* bk = (const float*)d_in[5];
    const float* wv = (const float*)d_in[6];
    const float* bv = (const float*)d_in[7];

    float* out_updated = (float*)d_out;                                   // [B,N]
    float* out_attn    = out_updated + (size_t)B_ROWS * N_COLS;           // [B,50,50]

    float* tk1  = (float*)d_ws;                                          // [B,50]
    float* tk2  = tk1 + (size_t)B_ROWS * K_TOP;                          // [B,50]
    int*   idx2 = (int*)(tk2 + (size_t)B_ROWS * K_TOP);                  // [B,50]

    // 1) zero the scatter target (B*N floats, 16B stores)
    {
        long long n4 = ((long long)B_ROWS * N_COLS) / 4;
        int threads = 256;
        long long blocks = (n4 + threads - 1) / threads;
        zero_f4_kernel<<<(unsigned)blocks, threads, 0, stream>>>((float4*)d_out, n4);
    }

    // 2) per-row top-k for A1 and A2 (one WGP-resident row each)
    {
        size_t smem_bytes = (size_t)(50016 + 32 + 32 + 4) * sizeof(float);
        topk_kernel<<<2 * B_ROWS, 1024, smem_bytes, stream>>>(A1, A2, tk1, tk2, idx2);
    }

    // 3) attention (WMMA scores + in-register softmax) and scatter
    attn_kernel<<<B_ROWS, 128, 0, stream>>>(tk1, tk2, idx2,
                                            wq, bq, wk, bk, wv, bv,
                                            out_updated, out_attn);

    (void)in_sizes; (void)n_in; (void)out_size; (void)ws_size;
}